// ConstituencyMFVI_89275190214780
// MI455X (gfx1250) — compile-verified
//
#include <hip/hip_runtime.h>
#include <hip/hip_bf16.h>
#include <stdint.h>

// ConstituencyMFVI on MI455X (gfx1250).
// One workgroup per (b,i): stage sp[b,i,:,:] (192x192 f32 = 147KB) into LDS via
// CDNA5 async global->LDS copies (ASYNCcnt path, th:TH_LOAD_NT streaming hint),
// then run the 3 sequential matvec iterations entirely out of LDS.
// HBM sees s_pair exactly once, and only the mask-live rows (~50% for triu mask).

#define NN      192
#define NITER   3
#define STRIDE  196                // 192 + 4 pad: 784B rows, 16B aligned, bank-spread
#define TPB     192                // 6 wave32s; 192 = 4 rows * 48 chunks -> clean mapping

__device__ __forceinline__ float fsigmoid(float x) {
    // 1/(1+exp(-x)) = 1/(1+exp2(-x*log2(e)))  -> v_exp_f32 + precise divide
    float e = __builtin_amdgcn_exp2f(-1.442695040888963f * x);
    return 1.0f / (1.0f + e);
}

__global__ __launch_bounds__(TPB)
void mfvi_kernel(const float* __restrict__ s_span,
                 const float* __restrict__ s_pair,
                 const unsigned char* __restrict__ mask,
                 float* __restrict__ out)
{
    __shared__ __align__(16) float sp[NN * STRIDE];   // 150,528 B
    __shared__ __align__(16) float sig[NN];
    __shared__ unsigned char msk[NN];

    const int bx = blockIdx.x;            // flat (b, i)
    const int b  = bx / NN;
    const int i  = bx - b * NN;
    const int j  = threadIdx.x;           // 0..191, owns output column j

    const size_t rowoff = (size_t)(b * NN + i) * NN;  // into [B,N,N]
    const float span_j  = s_span[rowoff + j];
    msk[j] = mask[rowoff + j];
    __syncthreads();

    // ---- Stage sp[b,i,:,:] into LDS: CDNA5 async global->LDS (ASYNCcnt) ----
    // Chunk c = 192*s + j  =>  row = j/48 + 4*s, t = j%48 (loop-invariant split).
    // Consecutive lanes take consecutive 16B chunks -> fully coalesced stream.
    // Mask-dead rows are zero-filled in LDS instead of fetched.
    {
        const uint64_t gbase = (uint64_t)(uintptr_t)(s_pair + rowoff * NN);
        const int r0 = j / 48;                 // 0..3
        const int t0 = j - r0 * 48;            // 0..47
        float*   dst  = &sp[r0 * STRIDE + t0 * 4];
        uint32_t goff = (uint32_t)j * 16u;     // byte offset of chunk j
        int      row  = r0;
        #pragma unroll 4
        for (int s = 0; s < 48; ++s) {
            if (msk[row]) {
                uint32_t ldsoff = (uint32_t)(uintptr_t)dst;  // low 32b of flat = LDS addr
                asm volatile(
                    "global_load_async_to_lds_b128 %0, %1, %2 offset:0 th:TH_LOAD_NT"
                    :: "v"(ldsoff), "v"(goff), "s"(gbase) : "memory");
            } else {
                *(float4*)dst = make_float4(0.f, 0.f, 0.f, 0.f);
            }
            dst  += 4 * STRIDE;      // +4 rows
            goff += TPB * 16u;       // +192 chunks
            row  += 4;
        }
    }
    asm volatile("s_wait_asynccnt 0" ::: "memory");
    __syncthreads();

    // ---- mask2o fixup: zero k == min(i,j) and k == max(i,j) in row j ----
    // (covered by the barrier inside the first iteration below)
    const int ls = (i < j) ? i : j;
    const int rs = (i < j) ? j : i;
    const bool live = (msk[j] != 0);
    if (live) {
        sp[j * STRIDE + ls] = 0.f;
        sp[j * STRIDE + rs] = 0.f;
    }

    // ---- 3 MFVI iterations: q_j = span_j + <SP_row_j , sigmoid(q)> ----
    float q = span_j;
    for (int it = 0; it < NITER; ++it) {
        sig[j] = fsigmoid(q);
        __syncthreads();               // covers fixup (it==0) + sig visibility
        if (live) {                    // dead rows: SP row is 0 -> q stays span_j
            float acc = span_j;
            const float4* row4 = (const float4*)&sp[j * STRIDE];  // 16B-aligned rows
            const float4* sig4 = (const float4*)sig;              // uniform broadcast
            #pragma unroll 8
            for (int t = 0; t < NN / 4; ++t) {
                float4 p = row4[t];
                float4 s = sig4[t];
                acc = fmaf(p.x, s.x, acc);
                acc = fmaf(p.y, s.y, acc);
                acc = fmaf(p.z, s.z, acc);
                acc = fmaf(p.w, s.w, acc);
            }
            q = acc;
        }
        __syncthreads();               // protect sig[] before next overwrite
    }

    out[rowoff + j] = fsigmoid(q);
}

extern "C" void kernel_launch(void* const* d_in, const int* in_sizes, int n_in,
                              void* d_out, int out_size, void* d_ws, size_t ws_size,
                              hipStream_t stream) {
    const float*         s_span = (const float*)d_in[0];
    const float*         s_pair = (const float*)d_in[1];
    const unsigned char* mask   = (const unsigned char*)d_in[2];  // jnp.bool_ = 1 byte
    float*               out    = (float*)d_out;

    const int B = in_sizes[0] / (NN * NN);   // 8
    dim3 grid(B * NN);                        // 1536 workgroups, one per (b,i)
    dim3 block(TPB);
    hipLaunchKernelGGL(mfvi_kernel, grid, block, 0, stream,
                       s_span, s_pair, mask, out);
    (void)n_in; (void)out_size; (void)d_ws; (void)ws_size;
}